// TopK_78752520339604
// MI455X (gfx1250) — compile-verified
//
#include <hip/hip_runtime.h>
#include <hip/hip_bf16.h>
#include <math.h>

// ---------------------------------------------------------------------------
// DeepSeek-style biased grouped top-k router for MI455X (gfx1250, wave32).
//
// Inputs : d_in[0] hidden_states [T,7168] f32 (UNUSED by the reference math)
//          d_in[1] router_logits [T,256]  f32
//          d_in[2] correction_bias [256]  f32
// Outputs (flat-concatenated in d_out):
//          [0      , T*8 )          topk_weights f32
//          [T*8    , T*16)          topk_ids     i32 (bit-stored)
//          [T*16   , T*16 + T*256)  router_logits f32 passthrough
//
// Memory-bound: ~33.6 MB total traffic -> ~1.4 us floor at 23.3 TB/s.
// One wave32 per token; all reductions via ds_bpermute butterflies;
// logits staged through LDS with gfx1250 async global->LDS loads.
// ---------------------------------------------------------------------------

#ifndef __has_builtin
#define __has_builtin(x) 0
#endif

#if defined(__HIP_DEVICE_COMPILE__) && defined(__gfx1250__) && \
    __has_builtin(__builtin_amdgcn_global_load_async_to_lds_b128)
#define USE_ASYNC_LDS 1
#else
#define USE_ASYNC_LDS 0
#endif

constexpr int   E_      = 256;  // experts
constexpr int   TOPK_   = 8;
constexpr int   TOPKG_  = 4;    // top groups kept (of 8 groups x 32 experts)
constexpr int   WPB     = 8;    // waves per block, 1 token per wave32
constexpr float SCALE_  = 2.5f; // routed scaling factor

#if USE_ASYNC_LDS
// Builtin signature (from hipcc diagnostic): both pointer args are
// int4-vector pointers; arg0 in AS1 (global), arg1 in AS3 (LDS).
typedef int v4i_t __attribute__((ext_vector_type(4)));
typedef __attribute__((address_space(1))) v4i_t* g_v4i_p;
typedef __attribute__((address_space(3))) v4i_t* l_v4i_p;
#endif

__device__ __forceinline__ float bperm_f(float v, int byte_idx) {
  return __int_as_float(__builtin_amdgcn_ds_bpermute(byte_idx, __float_as_int(v)));
}
__device__ __forceinline__ int bperm_i(int v, int byte_idx) {
  return __builtin_amdgcn_ds_bpermute(byte_idx, v);
}

__global__ __launch_bounds__(256, 4) void moe_router_topk_kernel(
    const float* __restrict__ logits,
    const float* __restrict__ bias,
    float* __restrict__ out_w,
    int*   __restrict__ out_ids,
    float* __restrict__ out_logits,
    int T)
{
  __shared__ float s_sig[WPB][E_];          // unbiased sigmoid scores
#if USE_ASYNC_LDS
  __shared__ float s_raw[WPB][E_];          // async-load staging for logits
#endif

  const int wv    = threadIdx.x >> 5;
  const int lane  = threadIdx.x & 31;
  const int token = blockIdx.x * WPB + wv;  // wave-uniform
  if (token >= T) return;
  const size_t tbase = (size_t)token * E_;

  // ---- load this token's 256 logits; lane owns experts [8*lane, 8*lane+8)
  float x[8];
#if USE_ASYNC_LDS
  {
    const float* g0 = logits + tbase + lane * 4;        // bytes [0,512)
    const float* g1 = logits + tbase + 128 + lane * 4;  // bytes [512,1024)
    __builtin_amdgcn_global_load_async_to_lds_b128(
        (g_v4i_p)(g0), (l_v4i_p)(&s_raw[wv][lane * 4]), 0, 0);
    __builtin_amdgcn_global_load_async_to_lds_b128(
        (g_v4i_p)(g1), (l_v4i_p)(&s_raw[wv][128 + lane * 4]), 0, 0);
    asm volatile("s_wait_asynccnt 0" ::: "memory");
    const float4 a = *(const float4*)&s_raw[wv][lane * 8];
    const float4 c = *(const float4*)&s_raw[wv][lane * 8 + 4];
    x[0]=a.x; x[1]=a.y; x[2]=a.z; x[3]=a.w;
    x[4]=c.x; x[5]=c.y; x[6]=c.z; x[7]=c.w;
  }
#else
  {
    const float4 a = *(const float4*)(logits + tbase + lane * 8);
    const float4 c = *(const float4*)(logits + tbase + lane * 8 + 4);
    x[0]=a.x; x[1]=a.y; x[2]=a.z; x[3]=a.w;
    x[4]=c.x; x[5]=c.y; x[6]=c.z; x[7]=c.w;
  }
#endif

  // ---- passthrough copy of router_logits (third output)
  *(float4*)(out_logits + tbase + lane * 8)     = make_float4(x[0],x[1],x[2],x[3]);
  *(float4*)(out_logits + tbase + lane * 8 + 4) = make_float4(x[4],x[5],x[6],x[7]);

  // ---- sigmoid + bias
  float cb[8];
  {
    const float4 c0 = *(const float4*)(bias + lane * 8);
    const float4 c1 = *(const float4*)(bias + lane * 8 + 4);
    cb[0]=c0.x; cb[1]=c0.y; cb[2]=c0.z; cb[3]=c0.w;
    cb[4]=c1.x; cb[5]=c1.y; cb[6]=c1.z; cb[7]=c1.w;
  }
  float s[8], b[8];
#pragma unroll
  for (int j = 0; j < 8; ++j) {
    s[j] = 1.0f / (1.0f + expf(-x[j]));
    b[j] = s[j] + cb[j];
  }
  // stash unbiased scores for the dynamic gather later
  *(float4*)&s_sig[wv][lane * 8]     = make_float4(s[0],s[1],s[2],s[3]);
  *(float4*)&s_sig[wv][lane * 8 + 4] = make_float4(s[4],s[5],s[6],s[7]);
  asm volatile("" ::: "memory");

  // ---- group score = sum of top-2 biased scores within group of 32
  // lane-local top-2 of its 8 values
  float m1 = b[0], m2 = -__builtin_inff();
#pragma unroll
  for (int j = 1; j < 8; ++j) {
    if (b[j] > m1) { m2 = m1; m1 = b[j]; }
    else           { m2 = fmaxf(m2, b[j]); }
  }
  // merge across the 4 lanes of the group (lane>>2 == group id)
#pragma unroll
  for (int m = 1; m <= 2; m <<= 1) {
    const int idx = (lane ^ m) << 2;
    const float om1 = bperm_f(m1, idx);
    const float om2 = bperm_f(m2, idx);
    const float nm1 = fmaxf(m1, om1);
    const float nm2 = fmaxf(fminf(m1, om1), fmaxf(m2, om2));
    m1 = nm1; m2 = nm2;
  }
  const float gscore = m1 + m2;     // uniform within the quad
  const int   g      = lane >> 2;

  // ---- top-4 groups via stable rank (ties -> lower group index)
  int rank = 0;
#pragma unroll
  for (int h = 0; h < 8; ++h) {
    const float o = bperm_f(gscore, (4 * h) << 2);  // broadcast group h's score
    rank += (o > gscore) || (o == gscore && h < g);
  }
  const bool sel = (rank < TOPKG_);

  // ---- mask: non-selected groups -> exactly 0.0 (reference semantics)
  float mv[8];
#pragma unroll
  for (int j = 0; j < 8; ++j) mv[j] = sel ? b[j] : 0.0f;

  // ---- stable top-8 over 256 masked values (ties -> lower expert index)
  float myw = 0.0f;
  int   myid = 0;
  const int base_e = lane << 3;
#pragma unroll
  for (int k = 0; k < TOPK_; ++k) {
    float bv = mv[0];
    int   bi = base_e;
#pragma unroll
    for (int j = 1; j < 8; ++j)
      if (mv[j] > bv) { bv = mv[j]; bi = base_e + j; }   // strict > keeps lowest idx
    // wave32 butterfly argmax (value desc, index asc)
#pragma unroll
    for (int m = 1; m < 32; m <<= 1) {
      const int idx = (lane ^ m) << 2;
      const float ov = bperm_f(bv, idx);
      const int   oi = bperm_i(bi, idx);
      if (ov > bv || (ov == bv && oi < bi)) { bv = ov; bi = oi; }
    }
    // lane k records the k-th winner; weight = UNbiased sigmoid score
    if (lane == k) { myid = bi; myw = s_sig[wv][bi]; }
    // owning lane retires the winner
    if ((bi >> 3) == lane) {
      const int slot = bi & 7;
#pragma unroll
      for (int j = 0; j < 8; ++j)
        if (j == slot) mv[j] = -__builtin_inff();
    }
  }

  // ---- normalize the 8 weights (held in lanes 0..7) and write out
  float wsum = (lane < TOPK_) ? myw : 0.0f;
#pragma unroll
  for (int m = 1; m <= 4; m <<= 1)
    wsum += bperm_f(wsum, (lane ^ m) << 2);
  if (lane < TOPK_) {
    out_w[(size_t)token * TOPK_ + lane]   = myw / wsum * SCALE_;
    out_ids[(size_t)token * TOPK_ + lane] = myid;
  }
}

extern "C" void kernel_launch(void* const* d_in, const int* in_sizes, int n_in,
                              void* d_out, int out_size, void* d_ws, size_t ws_size,
                              hipStream_t stream) {
  (void)n_in; (void)d_ws; (void)ws_size; (void)out_size;
  // d_in[0] = hidden_states (unused), d_in[1] = router_logits, d_in[2] = bias
  const float* logits = (const float*)d_in[1];
  const float* bias   = (const float*)d_in[2];
  const int T = in_sizes[1] / E_;

  float* out_w      = (float*)d_out;
  int*   out_ids    = (int*)((float*)d_out + (size_t)T * TOPK_);
  float* out_logits = (float*)d_out + (size_t)T * 2 * TOPK_;

  const int blocks = (T + WPB - 1) / WPB;
  moe_router_topk_kernel<<<dim3(blocks), dim3(256), 0, stream>>>(
      logits, bias, out_w, out_ids, out_logits, T);
}